// GNNEncoder_67035849556075
// MI455X (gfx1250) — compile-verified
//
#include <hip/hip_runtime.h>
#include <hip/hip_bf16.h>

#define HC 256          // H*C = 4*64
#define NEG_SLOPE 0.2f
#define BN_EPS 1e-5f

typedef __attribute__((ext_vector_type(2))) float    v2f;
typedef __attribute__((ext_vector_type(8))) float    v8f;
typedef __attribute__((ext_vector_type(4))) unsigned v4u;
typedef __attribute__((ext_vector_type(4))) int      v4i;
typedef __attribute__((ext_vector_type(8))) int      v8i;

// ---- swizzled-weight chunk geometry (per 32 K-rows of W) ----
// pair (W[k][n], W[k+1][n]) stored contiguously so a B fragment is ONE
// aligned ds_load_b64 (compiler fuses two into ds_load_2addr_b64).
// KPSTRIDE ≡ 16 (mod 32) pair-banks -> lanes 0-15 / 16-31 hit disjoint banks.
#define KPSTRIDE     272                 // float2 units
#define GSTRIDE      (2 * KPSTRIDE)      // 544 float2 per K-group of 4
#define CHUNK_PAIRS  (8 * GSTRIDE)       // 4352 float2
#define CHUNK_FLOATS (2 * CHUNK_PAIRS)   // 8704 floats
#define CHUNK_BYTES  (8 * CHUNK_PAIRS)   // 34816 bytes
#define WSW_FLOATS   (8 * CHUNK_FLOATS)  // full 256x256 W, swizzled+padded

// ---- order-preserving float<->uint map for atomic max on f32 ----
__device__ __forceinline__ unsigned f2ord(float f) {
  unsigned u = __float_as_uint(f);
  return (u & 0x80000000u) ? ~u : (u | 0x80000000u);
}
__device__ __forceinline__ float ord2f(unsigned o) {
  return (o & 0x80000000u) ? __uint_as_float(o & 0x7fffffffu)
                           : __uint_as_float(~o);
}

// ---- Tensor Data Mover: 1D copy of one 34816B swizzled chunk to LDS ----
#if defined(__clang_major__) && (__clang_major__ >= 23)
#define TDM_LOAD(g0, g1, z)                                                   \
  __builtin_amdgcn_tensor_load_to_lds((g0), (g1), (z), (z),                   \
                                      (v8i){0, 0, 0, 0, 0, 0, 0, 0}, 0)
#else
#define TDM_LOAD(g0, g1, z)                                                   \
  __builtin_amdgcn_tensor_load_to_lds((g0), (g1), (z), (z), 0)
#endif

__device__ __forceinline__ void tdm_load_chunk(unsigned ldsOff,
                                               const float* gsrc) {
  unsigned long long ga = (unsigned long long)(uintptr_t)gsrc;
  v4u g0;
  g0[0] = 1u;                                   // count=1, user descriptor
  g0[1] = ldsOff;                               // lds_addr (bytes)
  g0[2] = (unsigned)(ga & 0xffffffffu);         // global_addr[31:0]
  g0[3] = (unsigned)((ga >> 32) & 0x01ffffffu)  // global_addr[56:32]
          | (2u << 30);                         // type = 2 ("image")
  v8i g1;
  g1[0] = (int)(2u << 16);      // workgroup_mask=0 (not in cluster), 4B elems
  g1[1] = (int)(8704u << 16);   // tensor_dim0[15:0]  (1 row of 8704 dwords)
  g1[2] = (int)(1u << 16);      // tensor_dim0[31:16]=0, tensor_dim1=1
  g1[3] = (int)(8704u << 16);   // tile_dim0 = 8704
  g1[4] = 1;                    // tile_dim1 = 1, tile_dim2 = 0
  g1[5] = 8704;                 // tensor_dim0_stride[31:0]
  g1[6] = 0;                    // stride0[47:32]=0, stride1[15:0]=0
  g1[7] = 0;
  v4i z = {0, 0, 0, 0};
  TDM_LOAD(g0, g1, z);
}

// ============================================================================
// Pre-swizzle W[256x256] (row-major) into TDM-copyable fragment-pair chunks:
// Wsw pair[c*CHUNK_PAIRS + g*GSTRIDE + kp*KPSTRIDE + col]
//   = (W[(32c+4g+2kp)*256+col], W[(32c+4g+2kp+1)*256+col])
// ============================================================================
__global__ void __launch_bounds__(256) swizzle_w_kernel(
    const float* __restrict__ W, float2* __restrict__ Wsw) {
  int id = blockIdx.x * 256 + threadIdx.x;     // 32768 total
  int col = id & 255;
  int kp  = (id >> 8) & 1;
  int g   = (id >> 9) & 7;
  int c   = id >> 12;
  int k = c * 32 + g * 4 + kp * 2;
  float2 v;
  v.x = W[(size_t)k * HC + col];
  v.y = W[(size_t)(k + 1) * HC + col];
  Wsw[(size_t)c * CHUNK_PAIRS + g * GSTRIDE + kp * KPSTRIDE + col] = v;
}

// ============================================================================
// GEMM: Y[r,n] = sum_k X[r,k] * Wsw(k,n) + bias[n];  K = N_out = 256.
// 8 waves/block; wave = 16(M)x256(N) strip, 16 f32 WMMA accumulators.
// Double-buffered LDS: TDM DMA of chunk c+1 overlaps WMMA compute of chunk c
// (TENSORcnt in-order per wave => s_wait_tensorcnt 1 means chunk c landed).
// A/B fragments preloaded into register arrays so multiple ds_load_2addr_b64
// stay in flight instead of load->wait->wmma serialization.
// ============================================================================
__global__ void __launch_bounds__(256) gemm_bias_wmma_kernel(
    const float* __restrict__ X, const float* __restrict__ Wsw,
    const float* __restrict__ bias, float* __restrict__ Y, int nrows) {
  __shared__ __align__(16) float2 Bp[2][CHUNK_PAIRS];   // 2 x 34816 B
  const int tid  = threadIdx.x;
  const int wave = tid >> 5;
  const int lane = tid & 31;
  const int m    = lane & 15;                    // row (A) / col-in-tile (B,D)
  const int koff   = (lane < 16) ? 0 : 2;        // K-pair selector (ISA layout)
  const int kpBase = (lane < 16) ? 0 : KPSTRIDE;
  const int row0 = blockIdx.x * 128 + wave * 16;
  const bool rowValid = (row0 < nrows);

  v8f acc[16];
  v8f zero = {};
#pragma unroll
  for (int n = 0; n < 16; ++n) acc[n] = zero;

  int rA = rowValid ? (row0 + m) : 0;
  if (rA >= nrows) rA = nrows - 1;
  const float2* X2 = (const float2*)X;
  const size_t xbase = ((size_t)rA * HC) >> 1;             // float2 units
  const unsigned ldsBase = (unsigned)(uintptr_t)&Bp[0][0]; // LDS byte offset

  if (wave == 0) tdm_load_chunk(ldsBase, Wsw);             // chunk 0 -> buf 0

  for (int c = 0; c < 8; ++c) {                  // 8 chunks of 32 K-rows
    if (wave == 0) {
      if (c + 1 < 8) {                           // prefetch next chunk via TDM
        tdm_load_chunk(ldsBase + (unsigned)((c + 1) & 1) * CHUNK_BYTES,
                       Wsw + (size_t)(c + 1) * CHUNK_FLOATS);
        __builtin_amdgcn_s_wait_tensorcnt(1);    // in-order: chunk c complete
      } else {
        __builtin_amdgcn_s_wait_tensorcnt(0);
      }
    }
    __syncthreads();                             // publish chunk c to all waves

    const float2* buf = &Bp[c & 1][0];
    float2 afr[8];
#pragma unroll
    for (int g = 0; g < 8; ++g)                  // preload A fragments (chunk)
      afr[g] = X2[xbase + (size_t)((c * 32 + g * 4 + koff) >> 1)];

#pragma unroll
    for (int g = 0; g < 8; ++g) {                // K-steps of 4
      v2f a; a.x = afr[g].x; a.y = afr[g].y;
      const float2* frag = buf + g * GSTRIDE + kpBase + m;
      float2 bf[16];
#pragma unroll
      for (int n = 0; n < 16; ++n) bf[n] = frag[n * 16];   // 8x ds_load_2addr
#pragma unroll
      for (int n = 0; n < 16; ++n) {
        v2f b; b.x = bf[n].x; b.y = bf[n].y;
        acc[n] = __builtin_amdgcn_wmma_f32_16x16x4_f32(
            false, a, false, b, (short)0, acc[n], false, false);
      }
    }
    __syncthreads();          // all waves done with buf before it's re-DMA'd
  }

  if (rowValid) {
#pragma unroll
    for (int n = 0; n < 16; ++n) {
      const int col = n * 16 + m;
      const float bv = bias[col];
#pragma unroll
      for (int v = 0; v < 8; ++v) {
        int rr = (lane < 16) ? v : (v + 8);      // D layout: M split at lane 16
        int r = row0 + rr;
        if (r < nrows) Y[(size_t)r * HC + col] = acc[n][v] + bv;
      }
    }
  }
}

// ============================================================================
// Per-edge: e = leaky_relu(xl[src]+xr[dst]); score[e,h] = dot(e[h,:], att[h,:])
// then atomic segment-max into mOrd[dst,h]. One 256-thread block per edge.
// ============================================================================
__global__ void __launch_bounds__(256) edge_score_kernel(
    const float* __restrict__ xl, const float* __restrict__ xr,
    const int* __restrict__ src, const int* __restrict__ dst,
    const float* __restrict__ att, float* __restrict__ score,
    unsigned* __restrict__ mOrd) {
  const int e = blockIdx.x;
  const int t = threadIdx.x;               // channel = h*64 + c
  const int s = src[e], d = dst[e];
  float v = xl[(size_t)s * HC + t] + xr[(size_t)d * HC + t];
  v = (v > 0.0f) ? v : NEG_SLOPE * v;
  float p = v * att[t];
#pragma unroll
  for (int off = 16; off > 0; off >>= 1) p += __shfl_down(p, off, 32);
  __shared__ float partial[8];
  if ((t & 31) == 0) partial[t >> 5] = p;
  __syncthreads();
  if (t < 4) {                              // head h = t (64 ch = 2 waves)
    float sc = partial[2 * t] + partial[2 * t + 1];
    score[(size_t)e * 4 + t] = sc;
    atomicMax(&mOrd[(size_t)d * 4 + t], f2ord(sc));
  }
}

// a = exp(score - m[dst]); denom[dst] += a  (score overwritten with a)
__global__ void edge_exp_kernel(float* __restrict__ score,
                                const int* __restrict__ dst,
                                const unsigned* __restrict__ mOrd,
                                float* __restrict__ denom, long long e4) {
  long long i = (long long)blockIdx.x * blockDim.x + threadIdx.x;
  if (i >= e4) return;
  long long e = i >> 2;
  int h = (int)(i & 3);
  int d = dst[e];
  float mo = ord2f(mOrd[(size_t)d * 4 + h]);
  float a = expf(score[i] - mo);
  score[i] = a;
  atomicAdd(&denom[(size_t)d * 4 + h], a);
}

// pre[dst, t] += (a/denom[dst]) * xl[src, t]  — L2-resident atomics (51MB state)
__global__ void __launch_bounds__(256) edge_scatter_kernel(
    const float* __restrict__ xl, const float* __restrict__ aVal,
    const float* __restrict__ denom, const int* __restrict__ src,
    const int* __restrict__ dst, float* __restrict__ pre) {
  const int e = blockIdx.x;
  const int t = threadIdx.x;
  const int h = t >> 6;
  const int s = src[e], d = dst[e];
  float alpha = aVal[(size_t)e * 4 + h] / denom[(size_t)d * 4 + h];
  atomicAdd(&pre[(size_t)d * HC + t], alpha * xl[(size_t)s * HC + t]);
}

// ---- init helpers (graph-capture safe; no memset) ----
__global__ void init_seg_kernel(unsigned* __restrict__ mOrd,
                                float* __restrict__ denom,
                                float* __restrict__ sums, long long n4) {
  long long i = (long long)blockIdx.x * blockDim.x + threadIdx.x;
  if (i < n4) { mOrd[i] = 0x007fffffu /* ord(-inf) */; denom[i] = 0.0f; }
  if (i < 512) sums[i] = 0.0f;
}
__global__ void init_pre_kernel(float* __restrict__ pre,
                                const float* __restrict__ bo, long long total) {
  long long i = (long long)blockIdx.x * blockDim.x + threadIdx.x;
  if (i < total) pre[i] = bo[(int)(i & (HC - 1))];   // out = segsum + bo
}

// per-channel sum / sumsq over rows (coalesced, block = 128 rows)
__global__ void __launch_bounds__(256) bn_stats_kernel(
    const float* __restrict__ x, float* __restrict__ sums, int nrows) {
  const int ch = threadIdx.x;
  long long r0 = (long long)blockIdx.x * 128;
  long long r1 = r0 + 128; if (r1 > nrows) r1 = nrows;
  float s = 0.0f, s2 = 0.0f;
  for (long long r = r0; r < r1; ++r) {
    float v = x[r * HC + ch];
    s += v; s2 += v * v;
  }
  atomicAdd(&sums[ch], s);
  atomicAdd(&sums[HC + ch], s2);
}

__global__ void bn_relu_kernel(const float* __restrict__ x,
                               const float* __restrict__ sums,
                               const float* __restrict__ gamma,
                               const float* __restrict__ beta,
                               float* __restrict__ y, long long total,
                               float invN) {
  long long i = (long long)blockIdx.x * blockDim.x + threadIdx.x;
  if (i >= total) return;
  int ch = (int)(i & (HC - 1));
  float mu = sums[ch] * invN;
  float var = sums[HC + ch] * invN - mu * mu;
  float v = (x[i] - mu) * rsqrtf(var + BN_EPS) * gamma[ch] + beta[ch];
  y[i] = fmaxf(v, 0.0f);
}

// ============================================================================
extern "C" void kernel_launch(void* const* d_in, const int* in_sizes, int n_in,
                              void* d_out, int out_size, void* d_ws, size_t ws_size,
                              hipStream_t stream) {
  (void)n_in; (void)out_size; (void)ws_size;
  const float* x0 = (const float*)d_in[0];
  const int* ei   = (const int*)d_in[1];      // int32 per harness convention
  const int N      = in_sizes[0] / HC;        // 50000
  const long long E = in_sizes[1] / 2;        // 400000
  const int* src = ei;
  const int* dst = ei + E;

  // workspace: xl | xr | pre | score | mOrd | denom | sums | WswL | WswR
  float* ws = (float*)d_ws;
  const size_t NHC = (size_t)N * HC;
  float*    xl    = ws;
  float*    xr    = xl + NHC;
  float*    pre   = xr + NHC;
  float*    score = pre + NHC;
  unsigned* mOrd  = (unsigned*)(score + (size_t)E * 4);
  float*    denom = (float*)(mOrd + (size_t)N * 4);
  float*    sums  = denom + (size_t)N * 4;    // [sum(256) | sumsq(256)]
  float*    WswL  = sums + 512;
  float*    WswR  = WswL + WSW_FLOATS;

  dim3 blk(256);
  const int gemmGrid  = (N + 127) / 128;
  const long long e4  = E * 4;
  const int expGrid   = (int)((e4 + 255) / 256);
  const long long n4  = (long long)N * 4;
  const int segGrid   = (int)((n4 + 255) / 256);
  const long long tot = (long long)N * HC;
  const int elemGrid  = (int)((tot + 255) / 256);
  const float invN    = 1.0f / (float)N;

  const float* Xin = x0;
  for (int l = 0; l < 2; ++l) {
    const float* Wl    = (const float*)d_in[2 + 8 * l];
    const float* bl    = (const float*)d_in[3 + 8 * l];
    const float* Wr    = (const float*)d_in[4 + 8 * l];
    const float* br    = (const float*)d_in[5 + 8 * l];
    const float* att   = (const float*)d_in[6 + 8 * l];
    const float* bo    = (const float*)d_in[7 + 8 * l];
    const float* gamma = (const float*)d_in[8 + 8 * l];
    const float* beta  = (const float*)d_in[9 + 8 * l];

    swizzle_w_kernel<<<128, blk, 0, stream>>>(Wl, (float2*)WswL);
    swizzle_w_kernel<<<128, blk, 0, stream>>>(Wr, (float2*)WswR);
    gemm_bias_wmma_kernel<<<gemmGrid, blk, 0, stream>>>(Xin, WswL, bl, xl, N);
    gemm_bias_wmma_kernel<<<gemmGrid, blk, 0, stream>>>(Xin, WswR, br, xr, N);
    init_seg_kernel<<<segGrid, blk, 0, stream>>>(mOrd, denom, sums, n4);
    init_pre_kernel<<<elemGrid, blk, 0, stream>>>(pre, bo, tot);
    edge_score_kernel<<<(unsigned)E, blk, 0, stream>>>(xl, xr, src, dst, att,
                                                       score, mOrd);
    edge_exp_kernel<<<expGrid, blk, 0, stream>>>(score, dst, mOrd, denom, e4);
    edge_scatter_kernel<<<(unsigned)E, blk, 0, stream>>>(xl, score, denom, src,
                                                         dst, pre);
    bn_stats_kernel<<<gemmGrid, blk, 0, stream>>>(pre, sums, N);
    float* yout = (l == 0) ? pre : (float*)d_out;   // BN in-place on layer 1
    bn_relu_kernel<<<elemGrid, blk, 0, stream>>>(pre, sums, gamma, beta, yout,
                                                 tot, invN);
    Xin = pre;   // layer-2 input is BN'd layer-1 output
  }
}